// ViTBlockAdaLNZero_37477884625209
// MI455X (gfx1250) — compile-verified
//
#include <hip/hip_runtime.h>
#include <cstddef>

// ---------------------------------------------------------------------------
// ViT block with adaLN-Zero (DiT block), fp32, CDNA5 (gfx1250, wave32).
// All large GEMMs + attention use V_WMMA_F32_16X16X4_F32.
// GEMM staging uses GLOBAL_LOAD_ASYNC_TO_LDS_B128 (ASYNCcnt) + double buffer.
// B=8, N=1024, D=768, H=12, DH=64, MLP_HID=3072.
// ---------------------------------------------------------------------------

typedef __attribute__((ext_vector_type(2))) float v2f;
typedef __attribute__((ext_vector_type(8))) float v8f;
typedef __attribute__((ext_vector_type(4))) int   v4i;

#define WMMA_F32(a, b, c) \
  __builtin_amdgcn_wmma_f32_16x16x4_f32(false, (a), false, (b), (short)0, (c), false, false)

static constexpr int Bb      = 8;
static constexpr int Nn_seq  = 1024;
static constexpr int Dd      = 768;
static constexpr int Hh      = 12;
static constexpr int DH      = 64;
static constexpr int HALF    = 32;
static constexpr int ADA     = 6 * Dd;       // 4608
static constexpr int QKV_LD  = 3 * Dd;       // 2304
static constexpr int MROWS   = Bb * Nn_seq;  // 8192

// ---- CDNA5 async global->LDS copy (16B per lane), with portable fallback ---
__device__ __forceinline__ void async_copy16(const float* g, float* l) {
#if __has_builtin(__builtin_amdgcn_global_load_async_to_lds_b128)
  __builtin_amdgcn_global_load_async_to_lds_b128(
      (__attribute__((address_space(1))) v4i*)g,
      (__attribute__((address_space(3))) v4i*)l, 0, 0);
#else
  *(float4*)l = *(const float4*)g;
#endif
}

__device__ __forceinline__ void wait_async0() {
#if __has_builtin(__builtin_amdgcn_s_wait_asynccnt)
  __builtin_amdgcn_s_wait_asynccnt(0);
#else
  asm volatile("s_wait_asynccnt 0" ::: "memory");
#endif
}

// ---------------------------------------------------------------------------
// 1) ada = silu(c) @ ada_w + ada_b    (8 x 4608, K=768) — tiny, VALU kernel.
// ---------------------------------------------------------------------------
__global__ __launch_bounds__(256) void ada_kernel(const float* __restrict__ c,
                                                  const float* __restrict__ ada_w,
                                                  const float* __restrict__ ada_b,
                                                  float* __restrict__ ada) {
  const int tid = threadIdx.x;
  const int j   = blockIdx.x * 256 + tid;   // output column (0..4607)
  const int b   = blockIdx.y;               // batch row
  __shared__ float cs[Dd];
#pragma unroll
  for (int i = 0; i < 3; ++i) {
    float v = c[b * Dd + tid + i * 256];
    cs[tid + i * 256] = v / (1.0f + __expf(-v));   // silu
  }
  __syncthreads();
  float acc = ada_b[j];
#pragma unroll 4
  for (int k = 0; k < Dd; ++k) acc += cs[k] * ada_w[(size_t)k * ADA + j];
  ada[b * ADA + j] = acc;
}

// ---------------------------------------------------------------------------
// 2) h = layernorm(x) * (1 + scale) + shift     (one block per row of 768)
// ---------------------------------------------------------------------------
__global__ __launch_bounds__(256) void ln_mod_kernel(const float* __restrict__ x,
                                                     const float* __restrict__ ada,
                                                     int shift_off, int scale_off,
                                                     float* __restrict__ h) {
  const int row  = blockIdx.x;       // 0..8191
  const int b    = row >> 10;        // N = 1024
  const int tid  = threadIdx.x;
  const int lane = tid & 31;
  const int wave = tid >> 5;
  const float* xr = x + (size_t)row * Dd;

  float xa[3], s = 0.f, sq = 0.f;
#pragma unroll
  for (int i = 0; i < 3; ++i) {
    xa[i] = xr[tid + i * 256];
    s  += xa[i];
    sq += xa[i] * xa[i];
  }
#pragma unroll
  for (int o = 16; o > 0; o >>= 1) {
    s  += __shfl_xor(s, o, 32);
    sq += __shfl_xor(sq, o, 32);
  }
  __shared__ float s1[8], s2[8];
  if (lane == 0) { s1[wave] = s; s2[wave] = sq; }
  __syncthreads();
  s = 0.f; sq = 0.f;
#pragma unroll
  for (int w = 0; w < 8; ++w) { s += s1[w]; sq += s2[w]; }
  const float mu   = s * (1.0f / Dd);
  const float var  = sq * (1.0f / Dd) - mu * mu;
  const float rstd = rsqrtf(var + 1e-6f);

  const float* adab = ada + (size_t)b * ADA;
#pragma unroll
  for (int i = 0; i < 3; ++i) {
    const int d  = tid + i * 256;
    const float sc = adab[scale_off + d];
    const float sh = adab[shift_off + d];
    h[(size_t)row * Dd + d] = (xa[i] - mu) * rstd * (1.0f + sc) + sh;
  }
}

// ---------------------------------------------------------------------------
// 3) WMMA f32 GEMM: C[M x Nc] = A[M x K] * B[K x Nc] (+ epilogue)
//    Block tile 128x128, 8 waves = 4 row-groups x 2 col-groups; each wave
//    computes 32x64 (2 A-fragments x 4 (16x16) accumulator pairs).
//    K staged in steps of 32 through double-buffered LDS filled with
//    async global->LDS b128 copies (ASYNCcnt). Pads give conflict-free,
//    16B-aligned fragment reads (A row = 36 floats; B row = 136 floats:
//    +2 rows -> +16 banks, disjoint from the low lane-half).
// ---------------------------------------------------------------------------
enum { EPI_BIAS = 0, EPI_GELU = 1, EPI_RESGATE = 2 };

static constexpr int APAD = 36;    // 128 x 32 A tile, padded row
static constexpr int BPAD = 136;   // 32 x 128 B tile, padded row
static constexpr int ASZ  = 128 * APAD;
static constexpr int BSZ  = 32 * BPAD;

template <int EPI>
__global__ __launch_bounds__(256, 2) void gemm128x128(
    const float* __restrict__ A, const float* __restrict__ Bm,
    const float* __restrict__ bias, float* __restrict__ C,
    const float* __restrict__ R, const float* __restrict__ gate,
    int M, int Nc, int K) {
  const int tid  = threadIdx.x;
  const int lane = tid & 31;
  const int wave = tid >> 5;
  const int lr   = lane & 15;
  const int hi   = lane >> 4;
  const int wr   = wave >> 1;              // 0..3 row group (32 rows each)
  const int wc   = wave & 1;               // 0..1 col group (64 cols each)
  const int m0   = blockIdx.y * 128;
  const int n0   = blockIdx.x * 128;
  const int mw   = m0 + wr * 32;
  const int ncw  = wc * 64;

  __shared__ float As[2 * ASZ];
  __shared__ float Bs[2 * BSZ];

  // per-thread coop-load slots (A: 1024 float4, B: 1024 float4)
  const int ea_row[4] = {(tid + 0) >> 3, (tid + 256) >> 3, (tid + 512) >> 3, (tid + 768) >> 3};
  const int ea_kc     = (tid & 7) << 2;
  const int eb_row[4] = {(tid + 0) >> 5, (tid + 256) >> 5, (tid + 512) >> 5, (tid + 768) >> 5};
  const int eb_nc     = (tid & 31) << 2;

  const v8f zero = {0.f, 0.f, 0.f, 0.f, 0.f, 0.f, 0.f, 0.f};
  v8f acc[2][4] = {{zero, zero, zero, zero}, {zero, zero, zero, zero}};

  // issue tile kb into LDS buffer `buf` (async, 16B per lane per issue)
  auto issue_tile = [&](int kb, int buf) {
    float* as = As + buf * ASZ;
    float* bs = Bs + buf * BSZ;
#pragma unroll
    for (int i = 0; i < 4; ++i) {
      const int row = ea_row[i];
      async_copy16(A + (size_t)(m0 + row) * K + kb + ea_kc, &as[row * APAD + ea_kc]);
    }
#pragma unroll
    for (int i = 0; i < 4; ++i) {
      const int row = eb_row[i];
      async_copy16(Bm + (size_t)(kb + row) * Nc + n0 + eb_nc, &bs[row * BPAD + eb_nc]);
    }
  };

  issue_tile(0, 0);
  wait_async0();
  __syncthreads();

  for (int kb = 0; kb < K; kb += 32) {
    const int buf = (kb >> 5) & 1;
    if (kb + 32 < K) issue_tile(kb + 32, buf ^ 1);   // prefetch next tile

    const float* as = As + buf * ASZ;
    const float* bs = Bs + buf * BSZ;
#pragma unroll
    for (int kk = 0; kk < 32; kk += 4) {
      const v2f a0 = *(const v2f*)&as[(wr * 32 + lr) * APAD + kk + 2 * hi];
      const v2f a1 = *(const v2f*)&as[(wr * 32 + 16 + lr) * APAD + kk + 2 * hi];
#pragma unroll
      for (int t = 0; t < 4; ++t) {
        v2f bf;
        bf.x = bs[(kk + 2 * hi) * BPAD + ncw + t * 16 + lr];
        bf.y = bs[(kk + 2 * hi + 1) * BPAD + ncw + t * 16 + lr];
        acc[0][t] = WMMA_F32(a0, bf, acc[0][t]);
        acc[1][t] = WMMA_F32(a1, bf, acc[1][t]);
      }
    }
    wait_async0();        // next tile landed (this wave's copies)
    __syncthreads();      // all waves done reading `buf` and done copying
  }

  // ---- epilogue: lane holds col n, rows r(+8*hi) of each 16-row group ----
#pragma unroll
  for (int g = 0; g < 2; ++g) {
    const int mbase = mw + g * 16;
#pragma unroll
    for (int t = 0; t < 4; ++t) {
      const int n  = n0 + ncw + t * 16 + lr;
      const float bv = bias[n];
#pragma unroll
      for (int r = 0; r < 8; ++r) {
        const int m = mbase + r + 8 * hi;
        float v = acc[g][t][r] + bv;
        const size_t ci = (size_t)m * Nc + n;
        if (EPI == EPI_GELU) {
          v = 0.5f * v * (1.0f + erff(v * 0.70710678118f));   // exact gelu
        } else if (EPI == EPI_RESGATE) {
          v = R[ci] + gate[(size_t)(m >> 10) * ADA + n] * v;
        }
        C[ci] = v;
      }
    }
  }
}

// ---------------------------------------------------------------------------
// 4) RoPE on q and k inside the qkv buffer (B,N,3,H,DH)
// ---------------------------------------------------------------------------
__global__ __launch_bounds__(256) void rope_kernel(float* __restrict__ qkv,
                                                   const float* __restrict__ fcos,
                                                   const float* __restrict__ fsin) {
  int idx = blockIdx.x * 256 + threadIdx.x;     // B*N*2*H*HALF = 6,291,456
  const int i  = idx & 31; idx >>= 5;           // rotary pair index
  const int h  = idx % Hh; idx /= Hh;
  const int s  = idx & 1;  idx >>= 1;           // 0=q, 1=k
  const int n  = idx & 1023;
  const int b  = idx >> 10;
  const size_t base =
      ((size_t)(b * Nn_seq + n)) * QKV_LD + s * Dd + h * DH + 2 * i;
  const float x0 = qkv[base];
  const float x1 = qkv[base + 1];
  const float cv = fcos[n * HALF + i];
  const float sv = fsin[n * HALF + i];
  qkv[base]     = x0 * cv - x1 * sv;
  qkv[base + 1] = x0 * sv + x1 * cv;
}

// ---------------------------------------------------------------------------
// 5) Flash attention per (b, h, 128-query tile). Each wave owns 16 query
//    columns; computes S^T = K * Q^T tiles (softmax is per-lane over the 8
//    accumulator rows + one shfl_xor(16) pair-combine), then O^T = V^T * P.
// ---------------------------------------------------------------------------
__global__ __launch_bounds__(256, 2) void attn_kernel(const float* __restrict__ qkv,
                                                      float* __restrict__ o) {
  const int b    = blockIdx.z;
  const int h    = blockIdx.y;
  const int qt   = blockIdx.x;
  const int tid  = threadIdx.x;
  const int lane = tid & 31;
  const int wave = tid >> 5;
  const int lr   = lane & 15;
  const int hi   = lane >> 4;
  const int nq0  = qt * 128 + wave * 16;

  const float* qp = qkv + (size_t)b * Nn_seq * QKV_LD + h * DH;            // q
  const float* kp = qkv + (size_t)b * Nn_seq * QKV_LD + Dd + h * DH;       // k
  const float* vp = qkv + (size_t)b * Nn_seq * QKV_LD + 2 * Dd + h * DH;   // v

  // Pre-load Q as B-fragments (lane column = query row nq0+lr), pre-scaled.
  const float scale = 0.125f;            // DH^-0.5
  const int mq = nq0 + lr;
  v2f bq[16];
#pragma unroll
  for (int ks = 0; ks < 16; ++ks) {
    const int d0 = ks * 4 + 2 * hi;
    v2f t = *(const v2f*)&qp[(size_t)mq * QKV_LD + d0];
    bq[ks] = t * scale;
  }

  const v8f zero = {0.f, 0.f, 0.f, 0.f, 0.f, 0.f, 0.f, 0.f};
  v8f  oacc[4] = {zero, zero, zero, zero};
  float mrun = -1e30f, lrun = 0.f;

  __shared__ float Pl[8 * 16 * 17];      // per-wave 16x16 P scratch (pad 17)
  float* pw = &Pl[wave * 16 * 17];

  for (int nk0 = 0; nk0 < Nn_seq; nk0 += 16) {
    // ---- S^T tile (keys x queries) = K_tile * Q^T ----
    v8f st = zero;
    const int nk = nk0 + lr;             // A-fragment lane row = key row
#pragma unroll
    for (int ks = 0; ks < 16; ++ks) {
      const int d0 = ks * 4 + 2 * hi;
      const v2f ak = *(const v2f*)&kp[(size_t)nk * QKV_LD + d0];
      st = WMMA_F32(ak, bq[ks], st);
    }

    // ---- online softmax (per lane: 8 key rows of own query column) ----
    float tm = st[0];
#pragma unroll
    for (int r = 1; r < 8; ++r) tm = fmaxf(tm, st[r]);
    tm = fmaxf(tm, __shfl_xor(tm, 16, 32));
    const float mnew  = fmaxf(mrun, tm);
    const float alpha = __expf(mrun - mnew);
    float p[8], ts = 0.f;
#pragma unroll
    for (int r = 0; r < 8; ++r) { p[r] = __expf(st[r] - mnew); ts += p[r]; }
    ts += __shfl_xor(ts, 16, 32);
    lrun = lrun * alpha + ts;
    mrun = mnew;
#pragma unroll
    for (int dt = 0; dt < 4; ++dt)
#pragma unroll
      for (int r = 0; r < 8; ++r) oacc[dt][r] *= alpha;

    // ---- transpose P through per-wave LDS scratch ----
#pragma unroll
    for (int r = 0; r < 8; ++r) pw[(r + 8 * hi) * 17 + lr] = p[r];

    // ---- O^T += V^T * P  (A = V^T fragments from global, B = P from LDS) --
#pragma unroll
    for (int s = 0; s < 4; ++s) {
      v2f bp;
      bp.x = pw[(4 * s + 2 * hi) * 17 + lr];
      bp.y = pw[(4 * s + 2 * hi + 1) * 17 + lr];
      const float* vr = vp + (size_t)(nk0 + 4 * s + 2 * hi) * QKV_LD;
#pragma unroll
      for (int dt = 0; dt < 4; ++dt) {
        v2f av;
        av.x = vr[dt * 16 + lr];            // V[key][d] = V^T[d][key]
        av.y = vr[QKV_LD + dt * 16 + lr];   // next key row
        oacc[dt] = WMMA_F32(av, bp, oacc[dt]);
      }
    }
  }

  // ---- write O (lane: query col m = nq0+lr, rows d = dt*16 + r + 8*hi) ----
  const float inv = 1.0f / lrun;
#pragma unroll
  for (int dt = 0; dt < 4; ++dt)
#pragma unroll
    for (int r = 0; r < 8; ++r) {
      const int d = dt * 16 + r + 8 * hi;
      o[(size_t)(b * Nn_seq + nq0 + lr) * Dd + h * DH + d] = oacc[dt][r] * inv;
    }
}

// ---------------------------------------------------------------------------
// Launch
// ---------------------------------------------------------------------------
extern "C" void kernel_launch(void* const* d_in, const int* in_sizes, int n_in,
                              void* d_out, int out_size, void* d_ws, size_t ws_size,
                              hipStream_t stream) {
  const float* x      = (const float*)d_in[0];
  const float* c      = (const float*)d_in[1];
  const float* fcos   = (const float*)d_in[2];
  const float* fsin   = (const float*)d_in[3];
  const float* qkv_w  = (const float*)d_in[4];
  const float* qkv_b  = (const float*)d_in[5];
  const float* proj_w = (const float*)d_in[6];
  const float* proj_b = (const float*)d_in[7];
  const float* fc1_w  = (const float*)d_in[8];
  const float* fc1_b  = (const float*)d_in[9];
  const float* fc2_w  = (const float*)d_in[10];
  const float* fc2_b  = (const float*)d_in[11];
  const float* ada_w  = (const float*)d_in[12];
  const float* ada_b  = (const float*)d_in[13];
  float* out = (float*)d_out;

  // workspace layout (bytes):
  //   [ada 147456][x1 25165824][h/o/h2 25165824][qkv|m 100663296]
  char* ws = (char*)d_ws;
  float* wada = (float*)ws;                                      // 8 x 4608
  float* wx1  = (float*)(ws + 147456);                           // 8192 x 768
  float* wh   = (float*)(ws + 147456 + 25165824ull);             // 8192 x 768
  float* wbig = (float*)(ws + 147456 + 2ull * 25165824ull);      // 8192 x 3072

  // 1) adaLN projections
  ada_kernel<<<dim3(ADA / 256, Bb), 256, 0, stream>>>(c, ada_w, ada_b, wada);
  // 2) h1 = modulate(LN(x), sh_msa, sc_msa)
  ln_mod_kernel<<<MROWS, 256, 0, stream>>>(x, wada, 0, Dd, wh);
  // 3) qkv = h1 @ qkv_w + qkv_b
  gemm128x128<EPI_BIAS><<<dim3(QKV_LD / 128, MROWS / 128), 256, 0, stream>>>(
      wh, qkv_w, qkv_b, wbig, nullptr, nullptr, MROWS, QKV_LD, Dd);
  // 4) RoPE(q, k)
  rope_kernel<<<(Bb * Nn_seq * 2 * Hh * HALF) / 256, 256, 0, stream>>>(wbig, fcos, fsin);
  // 5) attention -> o (reuses wh)
  attn_kernel<<<dim3(Nn_seq / 128, Hh, Bb), 256, 0, stream>>>(wbig, wh);
  // 6) x1 = x + g_msa * (o @ proj_w + proj_b)
  gemm128x128<EPI_RESGATE><<<dim3(Dd / 128, MROWS / 128), 256, 0, stream>>>(
      wh, proj_w, proj_b, wx1, x, wada + 2 * Dd, MROWS, Dd, Dd);
  // 7) h2 = modulate(LN(x1), sh_mlp, sc_mlp)
  ln_mod_kernel<<<MROWS, 256, 0, stream>>>(wx1, wada, 3 * Dd, 4 * Dd, wh);
  // 8) m = gelu(h2 @ fc1_w + fc1_b)
  gemm128x128<EPI_GELU><<<dim3((4 * Dd) / 128, MROWS / 128), 256, 0, stream>>>(
      wh, fc1_w, fc1_b, wbig, nullptr, nullptr, MROWS, 4 * Dd, Dd);
  // 9) out = x1 + g_mlp * (m @ fc2_w + fc2_b)
  gemm128x128<EPI_RESGATE><<<dim3(Dd / 128, MROWS / 128), 256, 0, stream>>>(
      wbig, fc2_w, fc2_b, out, wx1, wada + 5 * Dd, MROWS, Dd, 4 * Dd);
}